// FastRadonTransform_85581518340113
// MI455X (gfx1250) — compile-verified
//
#include <hip/hip_runtime.h>

typedef __attribute__((ext_vector_type(2))) float v2f;
typedef __attribute__((ext_vector_type(8))) float v8f;

// Problem shape from the reference setup_inputs(): fixed.
constexpr int Bn = 2, Cn = 3, Hn = 384, Wn = 384, Kn = 180;

__device__ __forceinline__ float bilinear_sample(const float* __restrict__ img,
                                                 float fx, float fy) {
  // Emulates grid_sample(align_corners=True, bilinear, padding_mode='zeros'),
  // per-tap validity like the reference. Branchless so EXEC stays all-1s.
  const float x0f = floorf(fx);
  const float y0f = floorf(fy);
  const float wx1 = fx - x0f, wx0 = 1.0f - wx1;
  const float wy1 = fy - y0f, wy0 = 1.0f - wy1;
  const int x0 = (int)x0f, y0 = (int)y0f;
  const int x1 = x0 + 1,   y1 = y0 + 1;
  const int x0c = min(max(x0, 0), Wn - 1);
  const int x1c = min(max(x1, 0), Wn - 1);
  const int y0c = min(max(y0, 0), Hn - 1);
  const int y1c = min(max(y1, 0), Hn - 1);
  const float mx0 = (x0 >= 0 && x0 < Wn) ? 1.0f : 0.0f;
  const float mx1 = (x1 >= 0 && x1 < Wn) ? 1.0f : 0.0f;
  const float my0 = (y0 >= 0 && y0 < Hn) ? 1.0f : 0.0f;
  const float my1 = (y1 >= 0 && y1 < Hn) ? 1.0f : 0.0f;
  const float* r0 = img + (size_t)y0c * Wn;
  const float* r1 = img + (size_t)y1c * Wn;
  const float v00 = r0[x0c] * (my0 * mx0);
  const float v01 = r0[x1c] * (my0 * mx1);
  const float v10 = r1[x0c] * (my1 * mx0);
  const float v11 = r1[x1c] * (my1 * mx1);
  return wy0 * (wx0 * v00 + wx1 * v01) + wy1 * (wx0 * v10 + wx1 * v11);
}

__global__ void __launch_bounds__(96)
radon_wmma_kernel(const float* __restrict__ x,
                  const float* __restrict__ angles,
                  float* __restrict__ out) {
  const int htile = blockIdx.x;        // 0 .. Hn/16 - 1
  const int k     = blockIdx.y;        // angle index
  const int b     = blockIdx.z;        // batch
  const int wave  = (int)(threadIdx.x >> 5);   // channel (3 waves/block)
  const int lane  = (int)(threadIdx.x & 31);

  const int h0  = htile * 16;
  const int m   = lane & 15;           // A-matrix row (h within tile)
  const int klo = (lane >> 4) << 1;    // A-matrix K pair: lanes 0-15 -> K0/K1, 16-31 -> K2/K3

  const float theta = angles[b * Kn + k] * (3.14159265358979323846f / 180.0f);
  float s, c;
  sincosf(theta, &s, &c);

  const float invWm1 = 1.0f / (float)(Wn - 1);
  const float invHm1 = 1.0f / (float)(Hn - 1);
  const int   h    = h0 + m;
  const float ysh  = -1.0f + 2.0f * (float)h * invHm1;
  const float ys_s = ysh * s;
  const float ys_c = ysh * c;

  const float* img = x + (size_t)(b * Cn + wave) * Hn * Wn;

  v8f acc0 = {0.f, 0.f, 0.f, 0.f, 0.f, 0.f, 0.f, 0.f};
  v8f acc1 = {0.f, 0.f, 0.f, 0.f, 0.f, 0.f, 0.f, 0.f};
  v2f ones = {1.0f, 1.0f};   // B = 4x16 all-ones -> D[m,n] += sum_k A[m,k]

  // w-loop unrolled x2: 16 scattered gathers in flight per trip, two
  // independent WMMA accumulators (no C/D dependence between the pair).
  for (int wb = 0; wb < Wn; wb += 8) {
    v2f a0, a1;
#pragma unroll
    for (int j = 0; j < 2; ++j) {
      const int   wi  = wb + klo + j;
      const float xsw = -1.0f + 2.0f * (float)wi * invWm1;
      const float gx  =  xsw * c + ys_s;
      const float gy  =  ys_c - xsw * s;
      const float fx  = (gx + 1.0f) * (0.5f * (float)(Wn - 1));
      const float fy  = (gy + 1.0f) * (0.5f * (float)(Hn - 1));
      a0[j] = bilinear_sample(img, fx, fy);
    }
#pragma unroll
    for (int j = 0; j < 2; ++j) {
      const int   wi  = wb + 4 + klo + j;
      const float xsw = -1.0f + 2.0f * (float)wi * invWm1;
      const float gx  =  xsw * c + ys_s;
      const float gy  =  ys_c - xsw * s;
      const float fx  = (gx + 1.0f) * (0.5f * (float)(Wn - 1));
      const float fy  = (gy + 1.0f) * (0.5f * (float)(Hn - 1));
      a1[j] = bilinear_sample(img, fx, fy);
    }
    acc0 = __builtin_amdgcn_wmma_f32_16x16x4_f32(
        false, a0, false, ones, (short)0, acc0, false, false);
    acc1 = __builtin_amdgcn_wmma_f32_16x16x4_f32(
        false, a1, false, ones, (short)0, acc1, false, false);
  }

  const v8f acc = acc0 + acc1;

  // C/D layout: VGPR r, lane 0  -> D[M=r,   N=0]
  //             VGPR r, lane 16 -> D[M=8+r, N=0]
  if ((lane & 15) == 0) {
    const int mbase = (lane == 0) ? 0 : 8;
    float* o = out + ((size_t)(b * Cn + wave) * Hn + (h0 + mbase)) * Kn + k;
#pragma unroll
    for (int r = 0; r < 8; ++r) {
      o[(size_t)r * Kn] = acc[r];
    }
  }
}

extern "C" void kernel_launch(void* const* d_in, const int* in_sizes, int n_in,
                              void* d_out, int out_size, void* d_ws, size_t ws_size,
                              hipStream_t stream) {
  (void)in_sizes; (void)n_in; (void)out_size; (void)d_ws; (void)ws_size;
  const float* x      = (const float*)d_in[0];   // (B,C,H,W) f32
  const float* angles = (const float*)d_in[1];   // (B,K) f32
  float* out          = (float*)d_out;           // (B,C,H,K) f32

  dim3 grid(Hn / 16, Kn, Bn);  // (24, 180, 2)
  dim3 block(96);              // 3 waves: one per channel
  radon_wmma_kernel<<<grid, block, 0, stream>>>(x, angles, out);
}